// Flatten_2x2_13821204759252
// MI455X (gfx1250) — compile-verified
//
#include <hip/hip_runtime.h>

// Pure-permutation op: 2x2 block flatten of a (32, 3, 512, 512) fp32 tensor.
// 0 FLOPs, ~201 MB of HBM traffic -> bandwidth-bound (floor ~8.6 us @ 23.3 TB/s).
// Strategy: per-thread b128 loads from two adjacent input rows, register
// interleave, two contiguous b128 stores. Non-temporal hints since the
// touch-once working set (201 MB) exceeds the 192 MB L2.

typedef float f32x4 __attribute__((ext_vector_type(4)));

// Fixed problem dims from the reference.
#define DIM_W   512
#define DIM_HW  (512 * 512)
#define N_BC    (32 * 3)                 // batch * channels, contiguous layout
#define N_I     (DIM_W / 2)              // 256 output rows per (b,c) plane... (H/2)
#define N_JP    (DIM_W / 8)              // 64 "block-pair" columns per row
#define TOTAL_THREADS (N_BC * N_I * N_JP) // 1,572,864

__global__ __launch_bounds__(256) void Flatten2x2_kernel(
    const float* __restrict__ x, float* __restrict__ out) {
  // Thread coordinate decomposition (all powers of two):
  //   jp in [0,64): handles output cols [8*jp, 8*jp+8) == input cols [4*jp, 4*jp+4) of two rows
  //   i  in [0,256): output row i, sourced from input rows 2i and 2i+1
  //   bc in [0, 48): fused (b, c) plane index
  unsigned t  = blockIdx.x * blockDim.x + threadIdx.x;
  unsigned jp = t & (N_JP - 1);
  unsigned i  = (t >> 7) & (N_I - 1);
  unsigned bc = t >> 15;

  const float* src0 = x + (size_t)bc * DIM_HW + (size_t)(2u * i) * DIM_W + 4u * jp;

  // Two fully-coalesced 128-bit streaming loads (rows 2i and 2i+1).
  f32x4 r0 = __builtin_nontemporal_load((const f32x4*)src0);
  f32x4 r1 = __builtin_nontemporal_load((const f32x4*)(src0 + DIM_W));

  // Register interleave: out block j packs [x(2i,2j), x(2i,2j+1), x(2i+1,2j), x(2i+1,2j+1)].
  f32x4 o0 = { r0.x, r0.y, r1.x, r1.y };   // block j = 2*jp
  f32x4 o1 = { r0.z, r0.w, r1.z, r1.w };   // block j = 2*jp + 1

  float* dst = out + (size_t)bc * DIM_HW + (size_t)i * DIM_W + 8u * jp;

  // Two contiguous 128-bit streaming stores (1 KB contiguous per wave).
  __builtin_nontemporal_store(o0, (f32x4*)dst);
  __builtin_nontemporal_store(o1, (f32x4*)(dst + 4));
}

extern "C" void kernel_launch(void* const* d_in, const int* in_sizes, int n_in,
                              void* d_out, int out_size, void* d_ws, size_t ws_size,
                              hipStream_t stream) {
  const float* x  = (const float*)d_in[0];
  float* out      = (float*)d_out;

  const int block = 256;                       // 8 wave32 waves per workgroup
  const int grid  = TOTAL_THREADS / block;     // 6144 blocks
  Flatten2x2_kernel<<<grid, block, 0, stream>>>(x, out);
}